// LLMGNNRecommender_29592324670261
// MI455X (gfx1250) — compile-verified
//
#include <hip/hip_runtime.h>
#include <stdint.h>

#define NUM_USERS 100000
#define NUM_ITEMS 50000
#define NUM_NODES (NUM_USERS + NUM_ITEMS)
#define EMBED_DIM 64
#define N_LAYERS 3
#define NUM_EDGES 3200000
#define CHUNK 2048   // edges staged into LDS per block: 3 * 8KB = 24KB LDS

typedef unsigned int u32x4 __attribute__((ext_vector_type(4)));
typedef int i32x8 __attribute__((ext_vector_type(8)));
typedef int i32x4 __attribute__((ext_vector_type(4)));

#if __has_builtin(__builtin_amdgcn_tensor_load_to_lds)
#define HAVE_TDM 1
#else
#define HAVE_TDM 0
#endif

// ---------------------------------------------------------------------------
// elementwise helpers
// ---------------------------------------------------------------------------
__global__ void init_kernel(const float* __restrict__ user_e,
                            const float* __restrict__ item_e,
                            float* __restrict__ h,
                            float* __restrict__ acc) {
    size_t i = (size_t)blockIdx.x * blockDim.x + threadIdx.x;   // float4 index
    const size_t user4 = (size_t)NUM_USERS * EMBED_DIM / 4;
    const size_t tot4  = (size_t)NUM_NODES * EMBED_DIM / 4;
    if (i >= tot4) return;
    float4 v = (i < user4) ? ((const float4*)user_e)[i]
                           : ((const float4*)item_e)[i - user4];
    ((float4*)h)[i]   = v;
    ((float4*)acc)[i] = v;
}

__global__ void zero_kernel(float* __restrict__ p, int n4) {
    size_t i = (size_t)blockIdx.x * blockDim.x + threadIdx.x;
    if (i < (size_t)n4) ((float4*)p)[i] = make_float4(0.f, 0.f, 0.f, 0.f);
}

__global__ void accum_kernel(float* __restrict__ acc, const float* __restrict__ src, int n4) {
    size_t i = (size_t)blockIdx.x * blockDim.x + threadIdx.x;
    if (i >= (size_t)n4) return;
    float4 a = ((const float4*)acc)[i];
    float4 s = ((const float4*)src)[i];
    a.x += s.x; a.y += s.y; a.z += s.z; a.w += s.w;
    ((float4*)acc)[i] = a;
}

// final layer: acc = (acc + src) * 0.25  (folds the /(N_LAYERS+1) scale)
__global__ void accum_scale_kernel(float* __restrict__ acc, const float* __restrict__ src, int n4) {
    size_t i = (size_t)blockIdx.x * blockDim.x + threadIdx.x;
    if (i >= (size_t)n4) return;
    float4 a = ((const float4*)acc)[i];
    float4 s = ((const float4*)src)[i];
    a.x = (a.x + s.x) * 0.25f;
    a.y = (a.y + s.y) * 0.25f;
    a.z = (a.z + s.z) * 0.25f;
    a.w = (a.w + s.w) * 0.25f;
    ((float4*)acc)[i] = a;
}

// ---------------------------------------------------------------------------
// TDM: DMA a 1-D run of `nelem` 4-byte elements from global into LDS.
// Descriptor bitfields per CDNA5 ISA sec 8 (D# group 0/1); groups 2/3 unused
// (<= 2-D tensor). 6-arg builtin form (clang-23 / therock-10.0 headers).
// ---------------------------------------------------------------------------
#if HAVE_TDM
__device__ __forceinline__ void tdm_load_1d_b32(uint32_t lds_byte_addr,
                                                const void* gptr,
                                                uint32_t nelem) {
    uint64_t ga = (uint64_t)(uintptr_t)gptr;
    u32x4 g0;
    g0.x = 1u;                                    // count=1 valid descriptor, user mode
    g0.y = lds_byte_addr;                         // lds_addr [63:32]
    g0.z = (uint32_t)ga;                          // global_addr [95:64]
    g0.w = (uint32_t)((ga >> 32) & 0x01FFFFFFull) // global_addr [120:96]
           | (2u << 30);                          // type=2 ("image") [127:126]
    i32x8 g1;
    g1[0] = (int)(2u << 16);                      // wg_mask=0, data_size=2 (4B)
    g1[1] = (int)((nelem & 0xFFFFu) << 16);       // tensor_dim0[15:0] @ bits 63:48
    g1[2] = (int)((nelem >> 16) | (1u << 16));    // tensor_dim0[31:16], tensor_dim1=1
    g1[3] = (int)((nelem & 0xFFFFu) << 16);       // tile_dim0 = nelem (<=2048)
    g1[4] = 1;                                    // tile_dim1=1, tile_dim2=0 (unused)
    g1[5] = (int)nelem;                           // tensor_dim0_stride[31:0]
    g1[6] = 0;                                    // stride0[47:32], stride1[15:0]
    g1[7] = 0;
    i32x4 z4 = {0, 0, 0, 0};
    i32x8 z8 = {0, 0, 0, 0, 0, 0, 0, 0};
    __builtin_amdgcn_tensor_load_to_lds(g0, g1, z4, z4, z8, 0);
}

__device__ __forceinline__ uint32_t lds_offset_of(const void* generic_ptr) {
    // generic -> LDS(addrspace 3) cast; ptrtoint yields the 32-bit LDS byte offset
    return (uint32_t)(uintptr_t)(__attribute__((address_space(3))) const void*)generic_ptr;
}
#endif

// ---------------------------------------------------------------------------
// scatter: for each edge e (both directions), h_new[row] += w * h[col].
// One wave32 per edge; lane L owns dims {2L, 2L+1} (float2 -> global_load_b64,
// 256B coalesced per direction). Edge stream staged in LDS via TDM DMA.
// 2x unrolled so two edges' gathers are in flight before the atomic block.
// ---------------------------------------------------------------------------
__global__ void __launch_bounds__(256)
scatter_kernel(const float* __restrict__ h,
               float* __restrict__ hnew,
               const int* __restrict__ uidx,
               const int* __restrict__ iidx,
               const float* __restrict__ wval,
               int num_edges) {
    __shared__ int   s_u[CHUNK];
    __shared__ int   s_i[CHUNK];
    __shared__ float s_w[CHUNK];

    const int lane   = threadIdx.x & 31;
    const int wave   = threadIdx.x >> 5;
    const int nwaves = blockDim.x >> 5;

    const int base  = blockIdx.x * CHUNK;
    if (base >= num_edges) return;
    const int count = min(CHUNK, num_edges - base);

#if HAVE_TDM
    if (wave == 0) {   // one wave issues the three DMA descriptors (EXEC-independent op)
        tdm_load_1d_b32(lds_offset_of(s_u), uidx + base, (uint32_t)count);
        tdm_load_1d_b32(lds_offset_of(s_i), iidx + base, (uint32_t)count);
        tdm_load_1d_b32(lds_offset_of(s_w), wval + base, (uint32_t)count);
        __builtin_amdgcn_s_wait_tensorcnt(0);
    }
#else
    for (int t = threadIdx.x; t < count; t += blockDim.x) {
        s_u[t] = uidx[base + t];
        s_i[t] = iidx[base + t];
        s_w[t] = wval[base + t];
    }
#endif
    __syncthreads();

    int e = wave;
    for (; e + nwaves < count; e += 2 * nwaves) {
        const int e0 = e, e1 = e + nwaves;
        const int   u0 = s_u[e0];
        const int   u1 = s_u[e1];
        const int   v0 = s_i[e0] + NUM_USERS;
        const int   v1 = s_i[e1] + NUM_USERS;
        const float w0 = s_w[e0];
        const float w1 = s_w[e1];

        // four independent 256B-coalesced gathers in flight
        const float2 hu0 = ((const float2*)(h + (size_t)u0 * EMBED_DIM))[lane];
        const float2 hv0 = ((const float2*)(h + (size_t)v0 * EMBED_DIM))[lane];
        const float2 hu1 = ((const float2*)(h + (size_t)u1 * EMBED_DIM))[lane];
        const float2 hv1 = ((const float2*)(h + (size_t)v1 * EMBED_DIM))[lane];

        float* du0 = hnew + (size_t)u0 * EMBED_DIM + (size_t)lane * 2;
        float* dv0 = hnew + (size_t)v0 * EMBED_DIM + (size_t)lane * 2;
        float* du1 = hnew + (size_t)u1 * EMBED_DIM + (size_t)lane * 2;
        float* dv1 = hnew + (size_t)v1 * EMBED_DIM + (size_t)lane * 2;

        __hip_atomic_fetch_add(dv0 + 0, w0 * hu0.x, __ATOMIC_RELAXED, __HIP_MEMORY_SCOPE_AGENT);
        __hip_atomic_fetch_add(dv0 + 1, w0 * hu0.y, __ATOMIC_RELAXED, __HIP_MEMORY_SCOPE_AGENT);
        __hip_atomic_fetch_add(du0 + 0, w0 * hv0.x, __ATOMIC_RELAXED, __HIP_MEMORY_SCOPE_AGENT);
        __hip_atomic_fetch_add(du0 + 1, w0 * hv0.y, __ATOMIC_RELAXED, __HIP_MEMORY_SCOPE_AGENT);
        __hip_atomic_fetch_add(dv1 + 0, w1 * hu1.x, __ATOMIC_RELAXED, __HIP_MEMORY_SCOPE_AGENT);
        __hip_atomic_fetch_add(dv1 + 1, w1 * hu1.y, __ATOMIC_RELAXED, __HIP_MEMORY_SCOPE_AGENT);
        __hip_atomic_fetch_add(du1 + 0, w1 * hv1.x, __ATOMIC_RELAXED, __HIP_MEMORY_SCOPE_AGENT);
        __hip_atomic_fetch_add(du1 + 1, w1 * hv1.y, __ATOMIC_RELAXED, __HIP_MEMORY_SCOPE_AGENT);
    }
    if (e < count) {
        const int   u = s_u[e];
        const int   v = s_i[e] + NUM_USERS;
        const float w = s_w[e];

        const float2 hu = ((const float2*)(h + (size_t)u * EMBED_DIM))[lane];
        const float2 hv = ((const float2*)(h + (size_t)v * EMBED_DIM))[lane];

        float* du = hnew + (size_t)u * EMBED_DIM + (size_t)lane * 2;
        float* dv = hnew + (size_t)v * EMBED_DIM + (size_t)lane * 2;

        __hip_atomic_fetch_add(dv + 0, w * hu.x, __ATOMIC_RELAXED, __HIP_MEMORY_SCOPE_AGENT);
        __hip_atomic_fetch_add(dv + 1, w * hu.y, __ATOMIC_RELAXED, __HIP_MEMORY_SCOPE_AGENT);
        __hip_atomic_fetch_add(du + 0, w * hv.x, __ATOMIC_RELAXED, __HIP_MEMORY_SCOPE_AGENT);
        __hip_atomic_fetch_add(du + 1, w * hv.y, __ATOMIC_RELAXED, __HIP_MEMORY_SCOPE_AGENT);
    }
}

// ---------------------------------------------------------------------------
extern "C" void kernel_launch(void* const* d_in, const int* in_sizes, int n_in,
                              void* d_out, int out_size, void* d_ws, size_t ws_size,
                              hipStream_t stream) {
    const float* user_e = (const float*)d_in[0];
    const float* item_e = (const float*)d_in[1];
    const float* evals  = (const float*)d_in[2];
    const int*   uidx   = (const int*)d_in[3];
    const int*   iidx   = (const int*)d_in[4];

    float* acc  = (float*)d_out;                                   // 150000*64 floats
    float* h    = (float*)d_ws;                                    // ping
    float* hnew = (float*)d_ws + (size_t)NUM_NODES * EMBED_DIM;    // pong

    const int elem4 = NUM_NODES * EMBED_DIM / 4;     // 2.4M float4
    const int eb = 256;
    const int eg = (elem4 + eb - 1) / eb;
    const int nchunks = (NUM_EDGES + CHUNK - 1) / CHUNK;

    init_kernel<<<eg, eb, 0, stream>>>(user_e, item_e, h, acc);

    for (int l = 0; l < N_LAYERS; ++l) {
        zero_kernel<<<eg, eb, 0, stream>>>(hnew, elem4);
        scatter_kernel<<<nchunks, 256, 0, stream>>>(h, hnew, uidx, iidx, evals, NUM_EDGES);
        if (l < N_LAYERS - 1)
            accum_kernel<<<eg, eb, 0, stream>>>(acc, hnew, elem4);
        else
            accum_scale_kernel<<<eg, eb, 0, stream>>>(acc, hnew, elem4);
        float* t = h; h = hnew; hnew = t;
    }
}